// sLSTMCell_48112223650073
// MI455X (gfx1250) — compile-verified
//
#include <hip/hip_runtime.h>
#include <math.h>

typedef float v2f __attribute__((ext_vector_type(2)));
typedef float v8f __attribute__((ext_vector_type(8)));

#define HDIM 4096
#define IDIM 4096

// ---------------------------------------------------------------------------
// Kernel 1: all 8 matvecs -> 4 gate pre-activations, via V_WMMA_F32_16X16X4_F32
// grid.x = 4 gates * 256 row-blocks; block = 256 threads (8 wave32s)
// wave w in [0,4): K-slice of W_g @ x ; wave w in [4,8): K-slice of R_g @ h
// ---------------------------------------------------------------------------
__global__ __launch_bounds__(256)
void slstm_matvec_kernel(const float* __restrict__ x,
                         const float* __restrict__ h_prev,
                         const float* __restrict__ w_i, const float* __restrict__ w_f,
                         const float* __restrict__ w_o, const float* __restrict__ w_z,
                         const float* __restrict__ r_i, const float* __restrict__ r_f,
                         const float* __restrict__ r_o, const float* __restrict__ r_z,
                         const float* __restrict__ b_i, const float* __restrict__ b_f,
                         const float* __restrict__ b_o, const float* __restrict__ b_z,
                         float* __restrict__ pre)
{
    __shared__ float red[8][16];

    const int gate   = blockIdx.x >> 8;   // 0..3 : i, f, o, z
    const int rowblk = blockIdx.x & 255;  // 0..255 : 16-row block
    const int wave   = threadIdx.x >> 5;  // 0..7
    const int lane   = threadIdx.x & 31;

    const float* Wg = (gate == 0) ? w_i : (gate == 1) ? w_f : (gate == 2) ? w_o : w_z;
    const float* Rg = (gate == 0) ? r_i : (gate == 1) ? r_f : (gate == 2) ? r_o : r_z;
    const float* Bg = (gate == 0) ? b_i : (gate == 1) ? b_f : (gate == 2) ? b_o : b_z;

    const float* mat  = (wave < 4) ? Wg : Rg;        // uniform per wave
    const float* vec  = (wave < 4) ? x  : h_prev;
    const int    kbase = (wave & 3) * 1024;          // 1024-wide K slice per wave

    // A-matrix (16x4 fp32) lane layout: lanes 0-15 -> K=0,1 ; lanes 16-31 -> K=2,3
    const int row    = rowblk * 16 + (lane & 15);
    const int coloff = (lane >> 4) * 2;

    const float* rowp = mat + (size_t)row * IDIM + kbase + coloff;
    const float* vecp = vec + kbase + coloff;

    v8f acc = {};
    for (int k0 = 0; k0 < 1024; k0 += 32) {
        if (k0 + 128 < 1024)
            __builtin_prefetch(rowp + k0 + 128, 0, 0);   // global_prefetch_b8
#pragma unroll
        for (int kk = 0; kk < 32; kk += 4) {
            v2f a = *reinterpret_cast<const v2f*>(rowp + k0 + kk);  // weight tile rows
            v2f b = *reinterpret_cast<const v2f*>(vecp + k0 + kk);  // x/h chunk (bcast)
            // D = A x B + C  (all 16 columns of D identical == matvec partial)
            acc = __builtin_amdgcn_wmma_f32_16x16x4_f32(
                /*neg_a=*/false, a, /*neg_b=*/false, b,
                /*c_mod=*/(short)0, acc, /*reuse_a=*/false, /*reuse_b=*/false);
        }
    }

    // Column 0 of D: lane 0 holds rows 0..7 (VGPR r -> M=r), lane 16 holds rows 8..15
    if (lane == 0) {
#pragma unroll
        for (int r = 0; r < 8; ++r) red[wave][r] = acc[r];
    } else if (lane == 16) {
#pragma unroll
        for (int r = 0; r < 8; ++r) red[wave][8 + r] = acc[r];
    }
    __syncthreads();

    if (threadIdx.x < 16) {
        float s = 0.0f;
#pragma unroll
        for (int w2 = 0; w2 < 8; ++w2) s += red[w2][threadIdx.x];
        const int grow = rowblk * 16 + threadIdx.x;
        pre[gate * HDIM + grow] = s + Bg[grow];
    }
}

// ---------------------------------------------------------------------------
// Kernel 2: stabilized exponential gating (elementwise over H)
// out = [h_t | c_t | n_t | m_t]
// ---------------------------------------------------------------------------
__global__ __launch_bounds__(256)
void slstm_gate_kernel(const float* __restrict__ pre,
                       const float* __restrict__ c_prev,
                       const float* __restrict__ n_prev,
                       const float* __restrict__ m_prev,
                       float* __restrict__ out)
{
    const int j = blockIdx.x * blockDim.x + threadIdx.x;
    if (j >= HDIM) return;

    const float it = pre[0 * HDIM + j];
    const float ft = pre[1 * HDIM + j];
    const float ot = pre[2 * HDIM + j];
    const float zt = pre[3 * HDIM + j];

    // log_sigmoid(f) = min(f, 0) - log1p(exp(-|f|))   (stable)
    const float log_f = fminf(ft, 0.0f) - log1pf(expf(-fabsf(ft)));
    const float mp    = m_prev[j];
    const float mt    = fmaxf(log_f + mp, it);
    const float ip    = expf(it - mt);
    const float fp    = expf(log_f + mp - mt);

    const float ct = fp * c_prev[j] + ip * tanhf(zt);
    const float nt = fp * n_prev[j] + ip;
    const float sig_o = 1.0f / (1.0f + expf(-ot));
    const float ht = sig_o * tanhf(ct / nt);

    out[0 * HDIM + j] = ht;
    out[1 * HDIM + j] = ct;
    out[2 * HDIM + j] = nt;
    out[3 * HDIM + j] = mt;
}

// ---------------------------------------------------------------------------
extern "C" void kernel_launch(void* const* d_in, const int* in_sizes, int n_in,
                              void* d_out, int out_size, void* d_ws, size_t ws_size,
                              hipStream_t stream)
{
    const float* x      = (const float*)d_in[0];
    const float* h_prev = (const float*)d_in[1];
    const float* c_prev = (const float*)d_in[2];
    const float* n_prev = (const float*)d_in[3];
    const float* m_prev = (const float*)d_in[4];
    const float* w_i    = (const float*)d_in[5];
    const float* w_f    = (const float*)d_in[6];
    const float* w_o    = (const float*)d_in[7];
    const float* w_z    = (const float*)d_in[8];
    const float* r_i    = (const float*)d_in[9];
    const float* r_f    = (const float*)d_in[10];
    const float* r_o    = (const float*)d_in[11];
    const float* r_z    = (const float*)d_in[12];
    const float* b_i    = (const float*)d_in[13];
    const float* b_f    = (const float*)d_in[14];
    const float* b_o    = (const float*)d_in[15];
    const float* b_z    = (const float*)d_in[16];

    float* pre = (float*)d_ws;            // 4 * 4096 floats of scratch
    float* out = (float*)d_out;           // [h_t | c_t | n_t | m_t]

    slstm_matvec_kernel<<<dim3(4 * 256), dim3(256), 0, stream>>>(
        x, h_prev,
        w_i, w_f, w_o, w_z,
        r_i, r_f, r_o, r_z,
        b_i, b_f, b_o, b_z,
        pre);

    slstm_gate_kernel<<<dim3(HDIM / 256), dim3(256), 0, stream>>>(
        pre, c_prev, n_prev, m_prev, out);
}